// MjCambrianOptics_75874892251260
// MI455X (gfx1250) — compile-verified
//
#include <hip/hip_runtime.h>
#include <math.h>

// ---------------------------------------------------------------------------
// MjCambrianOptics on MI455X (gfx1250):
//   phase 1: PSF via angular-spectrum + batched COMPLEX GEMM (f32 WMMA 16x16x4,
//            LDS-staged 64x64 macro-tiles; 8 wave32/block, 16x32 per wave so
//            accumulators (8 x v8f = 64 VGPRs) never spill)
//   phase 2: FFT conv (Bluestein any-length DFT over power-of-2 LDS FFTs)
// ---------------------------------------------------------------------------

#define MXN      1023          // grid size
#define NPAD     1024          // padded stride for GEMM / transposes
#define CONV_N   2045          // conv FFT logical size (1023+1023-1)
#define CONV_PAD 2048          // padded stride
#define LOGM1    11            // Bluestein inner FFT for N=1023 -> 2048
#define LOGM2    12            // Bluestein inner FFT for N=2045 -> 4096
#define M1       2048
#define M2       4096
#define RES      511
#define KB       16            // GEMM K-panel depth

typedef float v2f __attribute__((ext_vector_type(2)));
typedef float v8f __attribute__((ext_vector_type(8)));

__device__ __forceinline__ float2 cmulf(float2 a, float2 b) {
  return make_float2(a.x * b.x - a.y * b.y, a.x * b.y + a.y * b.x);
}

// ------------------------------- utilities ---------------------------------
__global__ void k_zero(float* p, size_t n) {
  size_t i = (size_t)blockIdx.x * blockDim.x + threadIdx.x;
  size_t st = (size_t)gridDim.x * blockDim.x;
  for (; i < n; i += st) p[i] = 0.0f;
}

__global__ void k_reduce_sum(const float* __restrict__ x, size_t n, float* out) {
  __shared__ float red[256];
  size_t i = (size_t)blockIdx.x * blockDim.x + threadIdx.x;
  size_t st = (size_t)gridDim.x * blockDim.x;
  float s = 0.0f;
  for (; i < n; i += st) s += x[i];
  red[threadIdx.x] = s;
  __syncthreads();
  for (int o = 128; o > 0; o >>= 1) {
    if ((int)threadIdx.x < o) red[threadIdx.x] += red[threadIdx.x + o];
    __syncthreads();
  }
  if (threadIdx.x == 0) atomicAdd(out, red[0]);
}

// ------------------------------ Bluestein ----------------------------------
// chirp w[n] = exp(-i*pi*n^2/N), n^2 reduced mod 2N in integer for accuracy
__global__ void k_chirp(float2* w, int N) {
  int n = blockIdx.x * blockDim.x + threadIdx.x;
  if (n >= N) return;
  long long m = ((long long)n * (long long)n) % (2LL * (long long)N);
  double a = -3.14159265358979323846 * (double)m / (double)N;
  w[n] = make_float2((float)cos(a), (float)sin(a));
}

// b[m] = conj(w[|m|]) wrapped into length-M ring
__global__ void k_build_b(float2* b, const float2* __restrict__ w, int N, int M) {
  int m = blockIdx.x * blockDim.x + threadIdx.x;
  if (m >= M) return;
  float2 v = make_float2(0.0f, 0.0f);
  if (m < N)          { v = w[m];     v.y = -v.y; }
  else if (m > M - N) { v = w[M - m]; v.y = -v.y; }
  b[m] = v;
}

// In-LDS iterative radix-2 FFT of one length-M row per workgroup.
// dir=+1 forward, dir=-1 inverse (inverse includes 1/M scaling).
__global__ void k_fft_rows(float2* data, int logM, int rowStride, int dir) {
  extern __shared__ float2 smem[];
  const int M = 1 << logM;
  float2* row = data + (size_t)blockIdx.x * (size_t)rowStride;
  for (int i = threadIdx.x; i < M; i += blockDim.x) {
    unsigned r = __brev((unsigned)i) >> (32 - logM);
    smem[r] = row[i];
  }
  __syncthreads();
  const float sgn = (dir > 0) ? -1.0f : 1.0f;
  for (int len = 2; len <= M; len <<= 1) {
    const int half = len >> 1;
    const float ang0 = sgn * 2.0f * 3.14159265358979323846f / (float)len;
    for (int t = threadIdx.x; t < (M >> 1); t += blockDim.x) {
      const int blk = t / half, j = t - blk * half;
      const int i0 = blk * len + j, i1 = i0 + half;
      float s, c;
      __sincosf(ang0 * (float)j, &s, &c);
      float2 u = smem[i0], v = smem[i1];
      float2 w = make_float2(v.x * c - v.y * s, v.x * s + v.y * c);
      smem[i0] = make_float2(u.x + w.x, u.y + w.y);
      smem[i1] = make_float2(u.x - w.x, u.y - w.y);
    }
    __syncthreads();
  }
  const float scale = (dir > 0) ? 1.0f : (1.0f / (float)M);
  for (int i = threadIdx.x; i < M; i += blockDim.x) {
    float2 v = smem[i];
    row[i] = make_float2(v.x * scale, v.y * scale);
  }
}

__global__ void k_bs_pre(const float2* __restrict__ x, int xStride, float2* tmp,
                         int N, int M, const float2* __restrict__ w, int conjIn) {
  float2* trow = tmp + (size_t)blockIdx.x * (size_t)M;
  const float2* xrow = x + (size_t)blockIdx.x * (size_t)xStride;
  for (int n = threadIdx.x; n < M; n += blockDim.x) {
    float2 o = make_float2(0.0f, 0.0f);
    if (n < N) {
      float2 v = xrow[n];
      if (conjIn) v.y = -v.y;
      o = cmulf(v, w[n]);
    }
    trow[n] = o;
  }
}

__global__ void k_bs_mul(float2* tmp, const float2* __restrict__ Bf, int M) {
  float2* trow = tmp + (size_t)blockIdx.x * (size_t)M;
  for (int m = threadIdx.x; m < M; m += blockDim.x)
    trow[m] = cmulf(trow[m], Bf[m]);
}

__global__ void k_bs_post(const float2* __restrict__ tmp, float2* y, int yStride,
                          int N, int M, const float2* __restrict__ w, int conjOut,
                          float scale) {
  const float2* trow = tmp + (size_t)blockIdx.x * (size_t)M;
  float2* yrow = y + (size_t)blockIdx.x * (size_t)yStride;
  for (int k = threadIdx.x; k < N; k += blockDim.x) {
    float2 v = cmulf(trow[k], w[k]);
    if (conjOut) v.y = -v.y;
    yrow[k] = make_float2(v.x * scale, v.y * scale);
  }
}

// 32x32 LDS-tiled complex transpose (n multiple of 32)
__global__ void k_transpose(const float2* __restrict__ in, float2* __restrict__ out, int n) {
  __shared__ float2 tile[32][33];
  const int bx = blockIdx.x * 32, by = blockIdx.y * 32;
  const int x = bx + threadIdx.x;
  for (int yo = threadIdx.y; yo < 32; yo += 8)
    tile[yo][threadIdx.x] = in[(size_t)(by + yo) * n + x];
  __syncthreads();
  const int x2 = by + threadIdx.x;
  for (int yo = threadIdx.y; yo < 32; yo += 8)
    out[(size_t)(bx + yo) * n + x2] = tile[threadIdx.x][yo];
}

// --------------------------- optics field builders -------------------------
// fftshift folded: value at shifted source index (i+512)%1023
__global__ void k_build_u2(float2* u2, const float* __restrict__ dsum,
                           float invCnt, float lam) {
  int j = blockIdx.x * 16 + threadIdx.x;
  int i = blockIdx.y * 16 + threadIdx.y;
  if (i >= MXN || j >= MXN) return;
  int o = (i + 512) % MXN, p = (j + 512) % MXN;
  const float L = 0.01f;
  float x = -0.5f * L + (float)o * (L / (float)(MXN - 1));
  float y = -0.5f * L + (float)p * (L / (float)(MXN - 1));
  float d = dsum[0] * invCnt;
  float r2 = x * x + y * y;
  float A = (sqrtf(r2) / (0.0025f + 1e-7f) <= 1.0f) ? 1.0f : 0.0f;
  float k = 6.28318530717958647692f / lam;
  float ph = k * sqrtf(r2 + d * d);
  float s, c;
  __sincosf(ph, &s, &c);
  u2[(size_t)i * NPAD + j] = make_float2(A * c, A * s);
}

__global__ void k_build_H(float2* H, const float* __restrict__ dsum,
                          float invCnt, float lam) {
  int j = blockIdx.x * 16 + threadIdx.x;
  int i = blockIdx.y * 16 + threadIdx.y;
  if (i >= MXN || j >= MXN) return;
  int o = (i + 512) % MXN, p = (j + 512) % MXN;
  const float dx = 0.01f / 1023.0f;
  const float fmax = 1.0f / (2.0f * dx);
  float fx = -fmax + (float)o * (2.0f * fmax / (float)(MXN - 1));
  float fy = -fmax + (float)p * (2.0f * fmax / (float)(MXN - 1));
  float lf2 = (lam * fx) * (lam * fx) + (lam * fy) * (lam * fy);
  float2 v = make_float2(0.0f, 0.0f);
  if (lf2 < 1.0f) {
    float d = dsum[0] * invCnt;
    float k = 6.28318530717958647692f / lam;
    float ph = k * d * sqrtf(1.0f - lf2);
    float s, c;
    __sincosf(ph, &s, &c);
    v = make_float2(c, s);
  }
  H[(size_t)i * NPAD + j] = v;
}

// --------------------------- SoA split / merge -----------------------------
__global__ void k_split(const float2* __restrict__ src, float* re, float* im, int n) {
  int j = blockIdx.x * blockDim.x + threadIdx.x;
  int i = blockIdx.y;
  if (j >= n) return;
  float2 v = src[(size_t)i * n + j];
  re[(size_t)i * n + j] = v.x;
  im[(size_t)i * n + j] = v.y;
}

__global__ void k_merge(const float* __restrict__ re, const float* __restrict__ im,
                        float2* dst, int n) {
  int j = blockIdx.x * blockDim.x + threadIdx.x;
  int i = blockIdx.y;
  if (j >= n) return;
  dst[(size_t)i * n + j] = make_float2(re[(size_t)i * n + j], im[(size_t)i * n + j]);
}

// ------------------ complex GEMM via V_WMMA_F32_16X16X4_F32 ----------------
// C = A x B (complex), SoA planes, n x n (n multiple of 64).
// Workgroup = 8 wave32s (256 thr) computing a 64x64 macro-tile. K-panels of
// depth 16 staged in LDS (A 64x16 + B 16x64, re+im, ~17KB); each wave owns a
// 16x32 sub-tile (2 col-tiles), so complex accumulators are 8 x v8f = 64 VGPRs
// and never spill. 4 real WMMAs per k-step form the complex product.
// A-fragment layout per ISA (16x4 f32): lanes 0-15 {K0,K1}, lanes 16-31 {K2,K3}.
__global__ void k_cgemm_wmma(const float* __restrict__ Ar, const float* __restrict__ Ai,
                             const float* __restrict__ Br, const float* __restrict__ Bi,
                             float* __restrict__ Cr, float* __restrict__ Ci, int n) {
  __shared__ float As_r[64][KB + 1], As_i[64][KB + 1];   // +1 pad: kill bank conflicts
  __shared__ float Bs_r[KB][64], Bs_i[KB][64];
  const int lane = threadIdx.x;            // 0..31
  const int w    = threadIdx.y;            // 0..7 (wave in block)
  const int tid  = w * 32 + lane;          // 0..255
  const int half = lane >> 4;
  const int l    = lane & 15;
  const int strip   = w & 3;               // row strip (16 rows)
  const int colHalf = w >> 2;              // 0/1: left/right 32 columns
  const int r0   = blockIdx.y * 64;
  const int c0   = blockIdx.x * 64;

  const v8f vz = {0.f,0.f,0.f,0.f,0.f,0.f,0.f,0.f};
  v8f acRR[2], acII[2], acRI[2], acIR[2];
  for (int t = 0; t < 2; ++t) { acRR[t] = vz; acII[t] = vz; acRI[t] = vz; acIR[t] = vz; }

  for (int k0 = 0; k0 < n; k0 += KB) {
    {   // stage A panel (64 x KB) both planes: 1 x 4-float group per thread
      const int row = tid >> 2, c4 = (tid & 3) * 4;
      const float* pr = Ar + (size_t)(r0 + row) * n + k0 + c4;
      const float* pi = Ai + (size_t)(r0 + row) * n + k0 + c4;
      #pragma unroll
      for (int e = 0; e < 4; ++e) { As_r[row][c4 + e] = pr[e]; As_i[row][c4 + e] = pi[e]; }
    }
    {   // stage B panel (KB x 64) both planes
      const int krow = tid >> 4, c4 = (tid & 15) * 4;
      const float* pr = Br + (size_t)(k0 + krow) * n + c0 + c4;
      const float* pi = Bi + (size_t)(k0 + krow) * n + c0 + c4;
      #pragma unroll
      for (int e = 0; e < 4; ++e) { Bs_r[krow][c4 + e] = pr[e]; Bs_i[krow][c4 + e] = pi[e]; }
    }
    __syncthreads();
    if (k0 + KB < n) {   // pull next K-panel toward L2/L0 while WMMAs drain
      __builtin_prefetch(Ar + (size_t)(r0 + (tid >> 2)) * n + k0 + KB, 0, 0);
      __builtin_prefetch(Br + (size_t)(k0 + KB + (tid >> 4)) * n + c0, 0, 0);
    }
    const int rowL = strip * 16 + l;
    #pragma unroll
    for (int kk = 0; kk < KB; kk += 4) {
      const int ka = kk + half * 2;
      v2f ar, ai;
      ar.x = As_r[rowL][ka]; ar.y = As_r[rowL][ka + 1];
      ai.x = As_i[rowL][ka]; ai.y = As_i[rowL][ka + 1];
      #pragma unroll
      for (int tn = 0; tn < 2; ++tn) {
        const int cb = colHalf * 32 + tn * 16 + l;
        v2f br, bi;
        br.x = Bs_r[ka][cb]; br.y = Bs_r[ka + 1][cb];
        bi.x = Bs_i[ka][cb]; bi.y = Bs_i[ka + 1][cb];
        acRR[tn] = __builtin_amdgcn_wmma_f32_16x16x4_f32(false, ar, false, br, (short)0, acRR[tn], false, false);
        acII[tn] = __builtin_amdgcn_wmma_f32_16x16x4_f32(false, ai, false, bi, (short)0, acII[tn], false, false);
        acRI[tn] = __builtin_amdgcn_wmma_f32_16x16x4_f32(false, ar, false, bi, (short)0, acRI[tn], false, false);
        acIR[tn] = __builtin_amdgcn_wmma_f32_16x16x4_f32(false, ai, false, br, (short)0, acIR[tn], false, false);
      }
    }
    __syncthreads();
  }
  // D layout: VGPR v -> lanes 0-15: M=v, lanes 16-31: M=v+8; N = lane%16
  const int rw0 = r0 + strip * 16 + half * 8;
  #pragma unroll
  for (int tn = 0; tn < 2; ++tn) {
    const int cc = c0 + colHalf * 32 + tn * 16 + l;
    #pragma unroll
    for (int v = 0; v < 8; ++v) {
      size_t idx = (size_t)(rw0 + v) * n + cc;
      Cr[idx] = acRR[tn][v] - acII[tn][v];
      Ci[idx] = acRI[tn][v] + acIR[tn][v];
    }
  }
}

// ----------------------------- PSF magnitude -------------------------------
// ifftshift folded: psf[i][j] = |u[(i+511)%N][(j+511)%N]|^2 ; block-reduced sum
__global__ void k_psf(const float2* __restrict__ u, float* psf, float* ssum) {
  __shared__ float red[256];
  int j = blockIdx.x * 16 + threadIdx.x;
  int i = blockIdx.y * 16 + threadIdx.y;
  float m = 0.0f;
  if (i < MXN && j < MXN) {
    int o = (i + 511) % MXN, p = (j + 511) % MXN;
    float2 v = u[(size_t)o * NPAD + p];
    m = v.x * v.x + v.y * v.y;
    psf[(size_t)i * NPAD + j] = m;
  }
  int t = threadIdx.y * 16 + threadIdx.x;
  red[t] = m;
  __syncthreads();
  for (int o2 = 128; o2 > 0; o2 >>= 1) {
    if (t < o2) red[t] += red[t + o2];
    __syncthreads();
  }
  if (t == 0) atomicAdd(ssum, red[0]);
}

// --------------------------- convolution stage -----------------------------
__global__ void k_embed_img(const float* __restrict__ img, float2* dst) {
  int j = blockIdx.x * 16 + threadIdx.x;
  int i = blockIdx.y * 16 + threadIdx.y;
  if (i >= MXN || j >= MXN) return;
  dst[(size_t)i * CONV_PAD + j] = make_float2(img[(size_t)i * MXN + j], 0.0f);
}

// flipped kernel (correlation) with global PSF normalization folded in
__global__ void k_embed_ker(const float* __restrict__ psf, const float* __restrict__ ssum,
                            float2* dst) {
  int j = blockIdx.x * 16 + threadIdx.x;
  int i = blockIdx.y * 16 + threadIdx.y;
  if (i >= MXN || j >= MXN) return;
  float invS = 1.0f / (ssum[0] + 1e-7f);
  float v = psf[(size_t)(MXN - 1 - i) * NPAD + (MXN - 1 - j)] * invS;
  dst[(size_t)i * CONV_PAD + j] = make_float2(v, 0.0f);
}

__global__ void k_cmul_mat(float2* a, const float2* __restrict__ b, size_t n) {
  size_t i = (size_t)blockIdx.x * blockDim.x + threadIdx.x;
  size_t st = (size_t)gridDim.x * blockDim.x;
  for (; i < n; i += st) a[i] = cmulf(a[i], b[i]);
}

// full[sh+top+i][sw+left+j] with sh=sw=511, top=left=256 -> offset 767; clip
__global__ void k_crop(const float2* __restrict__ f, float* out, int c) {
  int j = blockIdx.x * 16 + threadIdx.x;
  int i = blockIdx.y * 16 + threadIdx.y;
  if (i >= RES || j >= RES) return;
  float v = f[(size_t)(767 + i) * CONV_PAD + (767 + j)].x;
  out[(size_t)c * RES * RES + (size_t)i * RES + j] = fminf(fmaxf(v, 0.0f), 1.0f);
}

// ------------------------------- launcher ----------------------------------
extern "C" void kernel_launch(void* const* d_in, const int* in_sizes, int n_in,
                              void* d_out, int out_size, void* d_ws, size_t ws_size,
                              hipStream_t stream) {
  (void)in_sizes; (void)n_in; (void)out_size; (void)ws_size;
  const float* image = (const float*)d_in[0];   // (3,1023,1023) f32
  const float* depth = (const float*)d_in[1];   // (1023,1023)   f32
  float* out = (float*)d_out;                   // (3,511,511)   f32

  char* ws = (char*)d_ws;
  size_t off = 0;
  auto alloc = [&](size_t b) -> char* {
    char* p = ws + off;
    off = (off + b + 255) & ~(size_t)255;
    return p;
  };
  float2* W1  = (float2*)alloc((size_t)MXN * 8);
  float2* B1f = (float2*)alloc((size_t)M1 * 8);
  float2* W2  = (float2*)alloc((size_t)CONV_N * 8);
  float2* B2f = (float2*)alloc((size_t)M2 * 8);
  float*  SUM = (float*)alloc(64);                       // [0]=depth sum, [1]=psf sum
  float2* CA  = (float2*)alloc((size_t)NPAD * NPAD * 8); // u2 / fft2(u2) / result
  float2* CB  = (float2*)alloc((size_t)NPAD * NPAD * 8); // fftshift(H)
  float2* CT  = (float2*)alloc((size_t)NPAD * NPAD * 8); // transpose scratch
  float*  P[6];
  for (int t = 0; t < 6; ++t) P[t] = (float*)alloc((size_t)NPAD * NPAD * 4);
  float*  PSF = (float*)alloc((size_t)3 * NPAD * NPAD * 4);
  float2* TMP = (float2*)alloc((size_t)CONV_N * M2 * 8); // Bluestein work rows
  float2* FI  = (float2*)alloc((size_t)CONV_PAD * CONV_PAD * 8);
  float2* FK  = (float2*)alloc((size_t)CONV_PAD * CONV_PAD * 8);
  float2* FT  = (float2*)alloc((size_t)CONV_PAD * CONV_PAD * 8);

  const float lams[3] = {6.1e-07f, 5.3e-07f, 4.7e-07f};
  const float invCnt = 1.0f / ((float)MXN * (float)MXN);
  dim3 b256(256);
  dim3 g16((MXN + 15) / 16, (MXN + 15) / 16), blk16(16, 16);

  // Bluestein tables (per-launch; deterministic)
  k_chirp<<<(MXN + 255) / 256, b256, 0, stream>>>(W1, MXN);
  k_chirp<<<(CONV_N + 255) / 256, b256, 0, stream>>>(W2, CONV_N);
  k_build_b<<<(M1 + 255) / 256, b256, 0, stream>>>(B1f, W1, MXN, M1);
  k_fft_rows<<<1, 256, (size_t)M1 * sizeof(float2), stream>>>(B1f, LOGM1, M1, +1);
  k_build_b<<<(M2 + 255) / 256, b256, 0, stream>>>(B2f, W2, CONV_N, M2);
  k_fft_rows<<<1, 256, (size_t)M2 * sizeof(float2), stream>>>(B2f, LOGM2, M2, +1);

  k_zero<<<1, 256, 0, stream>>>(SUM, 16);
  k_reduce_sum<<<512, 256, 0, stream>>>(depth, (size_t)MXN * MXN, &SUM[0]);
  k_zero<<<2048, 256, 0, stream>>>(PSF, (size_t)3 * NPAD * NPAD);

  // Bluestein length-N DFT applied to `rows` rows of `mat` (row stride = stride)
  auto bs_rows = [&](float2* mat, int stride, int rows, int N, int logM,
                     const float2* w, const float2* Bf, int inverse) {
    const int M = 1 << logM;
    k_bs_pre<<<rows, b256, 0, stream>>>(mat, stride, TMP, N, M, w, inverse);
    k_fft_rows<<<rows, 256, (size_t)M * sizeof(float2), stream>>>(TMP, logM, M, +1);
    k_bs_mul<<<rows, b256, 0, stream>>>(TMP, Bf, M);
    k_fft_rows<<<rows, 256, (size_t)M * sizeof(float2), stream>>>(TMP, logM, M, -1);
    k_bs_post<<<rows, b256, 0, stream>>>(TMP, mat, stride, N, M, w, inverse,
                                         inverse ? 1.0f / (float)N : 1.0f);
  };
  auto fft2_1023 = [&](float2* mat, int inverse) {
    bs_rows(mat, NPAD, MXN, MXN, LOGM1, W1, B1f, inverse);
    k_transpose<<<dim3(NPAD / 32, NPAD / 32), dim3(32, 8), 0, stream>>>(mat, CT, NPAD);
    bs_rows(CT, NPAD, MXN, MXN, LOGM1, W1, B1f, inverse);
    k_transpose<<<dim3(NPAD / 32, NPAD / 32), dim3(32, 8), 0, stream>>>(CT, mat, NPAD);
  };
  auto fft2_2045 = [&](float2* mat, int inverse) {
    bs_rows(mat, CONV_PAD, CONV_N, CONV_N, LOGM2, W2, B2f, inverse);
    k_transpose<<<dim3(CONV_PAD / 32, CONV_PAD / 32), dim3(32, 8), 0, stream>>>(mat, FT, CONV_PAD);
    bs_rows(FT, CONV_PAD, CONV_N, CONV_N, LOGM2, W2, B2f, inverse);
    k_transpose<<<dim3(CONV_PAD / 32, CONV_PAD / 32), dim3(32, 8), 0, stream>>>(FT, mat, CONV_PAD);
  };

  // -------- phase 1: depth-invariant PSF per wavelength --------
  for (int c = 0; c < 3; ++c) {
    k_zero<<<1024, 256, 0, stream>>>((float*)CA, (size_t)NPAD * NPAD * 2);
    k_zero<<<1024, 256, 0, stream>>>((float*)CB, (size_t)NPAD * NPAD * 2);
    k_build_u2<<<g16, blk16, 0, stream>>>(CA, &SUM[0], invCnt, lams[c]);  // fftshift(u2)
    k_build_H<<<g16, blk16, 0, stream>>>(CB, &SUM[0], invCnt, lams[c]);   // fftshift(H)
    fft2_1023(CA, 0);                                                     // fft2(fftshift(u2))
    // complex GEMM: CB (A) x CA (B) via WMMA on zero-padded 1024x1024 planes
    k_split<<<dim3(NPAD / 256, NPAD), b256, 0, stream>>>(CB, P[0], P[1], NPAD);
    k_split<<<dim3(NPAD / 256, NPAD), b256, 0, stream>>>(CA, P[2], P[3], NPAD);
    k_cgemm_wmma<<<dim3(NPAD / 64, NPAD / 64), dim3(32, 8), 0, stream>>>(
        P[0], P[1], P[2], P[3], P[4], P[5], NPAD);
    k_merge<<<dim3(NPAD / 256, NPAD), b256, 0, stream>>>(P[4], P[5], CA, NPAD);
    fft2_1023(CA, 1);                                                     // ifft2
    k_psf<<<g16, blk16, 0, stream>>>(CA, PSF + (size_t)c * NPAD * NPAD, &SUM[1]);
  }

  // -------- phase 2: per-channel FFT correlation, crop, clip --------
  for (int c = 0; c < 3; ++c) {
    k_zero<<<2048, 256, 0, stream>>>((float*)FI, (size_t)CONV_PAD * CONV_PAD * 2);
    k_zero<<<2048, 256, 0, stream>>>((float*)FK, (size_t)CONV_PAD * CONV_PAD * 2);
    k_embed_img<<<g16, blk16, 0, stream>>>(image + (size_t)c * MXN * MXN, FI);
    k_embed_ker<<<g16, blk16, 0, stream>>>(PSF + (size_t)c * NPAD * NPAD, &SUM[1], FK);
    fft2_2045(FI, 0);
    fft2_2045(FK, 0);
    k_cmul_mat<<<2048, 256, 0, stream>>>(FI, FK, (size_t)CONV_PAD * CONV_PAD);
    fft2_2045(FI, 1);
    k_crop<<<dim3((RES + 15) / 16, (RES + 15) / 16), blk16, 0, stream>>>(FI, out, c);
  }
}